// GCNLayerONNX_51994874085832
// MI455X (gfx1250) — compile-verified
//
#include <hip/hip_runtime.h>

#define NUM_NODES 50000
#define NUM_EDGES 800000
#define CH 64

typedef __attribute__((ext_vector_type(2))) float v2f;
typedef __attribute__((ext_vector_type(8))) float v8f;

__global__ void zero_f32(float* __restrict__ p, int n) {
    int i = blockIdx.x * blockDim.x + threadIdx.x;
    int stride = gridDim.x * blockDim.x;
    for (; i < n; i += stride) p[i] = 0.0f;
}

__global__ void deg_kernel(const int* __restrict__ col, float* __restrict__ deg, int E) {
    int e = blockIdx.x * blockDim.x + threadIdx.x;
    if (e < E) atomicAdd(&deg[col[e]], 1.0f);
}

__global__ void dis_kernel(const float* __restrict__ deg, float* __restrict__ dis, int N) {
    int i = blockIdx.x * blockDim.x + threadIdx.x;
    if (i < N) {
        float d = deg[i];
        dis[i] = (d > 0.0f) ? rsqrtf(d + 1e-16f) : 0.0f;
    }
}

__global__ void norm_kernel(const int* __restrict__ row, const int* __restrict__ col,
                            const float* __restrict__ dis, float* __restrict__ norm, int E) {
    int e = blockIdx.x * blockDim.x + threadIdx.x;
    if (e < E) norm[e] = dis[row[e]] * dis[col[e]];
}

// One thread per (edge, group of 4 channels): float4 gather of x[row], 4 L2 atomic adds into agg[col].
__global__ void scatter_kernel(const int* __restrict__ row, const int* __restrict__ col,
                               const float* __restrict__ x, const float* __restrict__ norm,
                               float* __restrict__ agg, int E) {
    int tid = blockIdx.x * blockDim.x + threadIdx.x;
    int e = tid >> 4;            // 16 float4-groups per edge (64 channels)
    if (e >= E) return;
    int g = (tid & 15) << 2;
    int r = row[e];
    int c = col[e];
    float nv = norm[e];
    float4 v = *(const float4*)(x + r * CH + g);
    float* dst = agg + c * CH + g;
    atomicAdd(dst + 0, nv * v.x);
    atomicAdd(dst + 1, nv * v.y);
    atomicAdd(dst + 2, nv * v.z);
    atomicAdd(dst + 3, nv * v.w);
}

// out = relu(agg @ W + b), agg [M,64], W [64,64], via V_WMMA_F32_16X16X4_F32.
// Block = 128 threads = 4 waves; wave w owns N-tile w (cols w*16..w*16+15),
// blockIdx.x owns M-tile (rows blockIdx.x*16..+15). K=64 -> 16 WMMA steps.
__global__ void gemm_bias_relu_wmma(const float* __restrict__ agg,
                                    const float* __restrict__ W,
                                    const float* __restrict__ bias,
                                    float* __restrict__ out) {
    const int wave = threadIdx.x >> 5;
    const int lane = threadIdx.x & 31;
    const int half = lane >> 4;      // 0: lanes 0-15, 1: lanes 16-31
    const int l    = lane & 15;
    const int m0   = blockIdx.x * 16;
    const int n0   = wave * 16;

    v8f acc = {};
    #pragma unroll
    for (int k0 = 0; k0 < CH; k0 += 4) {
        const int kA = k0 + half * 2;   // lane halves hold K pairs (ISA 16x4 f32 A layout)
        v2f a, b;
        a.x = agg[(m0 + l) * CH + kA];
        a.y = agg[(m0 + l) * CH + kA + 1];
        b.x = W[kA * CH + n0 + l];        // B mirrored: lane = N column, pair = K
        b.y = W[(kA + 1) * CH + n0 + l];
        acc = __builtin_amdgcn_wmma_f32_16x16x4_f32(
            /*neg_a=*/false, a, /*neg_b=*/false, b,
            /*c_mod=*/(short)0, acc, /*reuse_a=*/false, /*reuse_b=*/false);
    }

    const float bv = bias[n0 + l];
    #pragma unroll
    for (int v = 0; v < 8; ++v) {
        const int r = m0 + half * 8 + v;   // C/D layout: VGPR v = row (half*8 + v)
        float val = acc[v] + bv;
        out[r * CH + n0 + l] = val > 0.0f ? val : 0.0f;
    }
}

extern "C" void kernel_launch(void* const* d_in, const int* in_sizes, int n_in,
                              void* d_out, int out_size, void* d_ws, size_t ws_size,
                              hipStream_t stream) {
    const float* x          = (const float*)d_in[0];
    const int*   edge_index = (const int*)d_in[1];   // [2, E] row-major
    const float* W          = (const float*)d_in[2];
    const float* b          = (const float*)d_in[3];
    const int*   row = edge_index;
    const int*   col = edge_index + NUM_EDGES;
    float* out = (float*)d_out;

    // Workspace layout
    float* deg  = (float*)d_ws;            // [NUM_NODES]
    float* dis  = deg + NUM_NODES;         // [NUM_NODES]
    float* norm = dis + NUM_NODES;         // [NUM_EDGES]
    float* agg  = norm + NUM_EDGES;        // [NUM_NODES * CH]

    // Zero accumulators every call (harness does not re-poison between replays)
    zero_f32<<<1024, 256, 0, stream>>>(deg, NUM_NODES);
    zero_f32<<<4096, 256, 0, stream>>>(agg, NUM_NODES * CH);

    deg_kernel<<<(NUM_EDGES + 255) / 256, 256, 0, stream>>>(col, deg, NUM_EDGES);
    dis_kernel<<<(NUM_NODES + 255) / 256, 256, 0, stream>>>(deg, dis, NUM_NODES);
    norm_kernel<<<(NUM_EDGES + 255) / 256, 256, 0, stream>>>(row, col, dis, norm, NUM_EDGES);

    const int scatter_threads = NUM_EDGES * 16;  // 16 float4-groups per edge
    scatter_kernel<<<(scatter_threads + 255) / 256, 256, 0, stream>>>(row, col, x, norm, agg, NUM_EDGES);

    gemm_bias_relu_wmma<<<NUM_NODES / 16, 128, 0, stream>>>(agg, W, b, out);
}